// SingleNeuronModel_19043884990565
// MI455X (gfx1250) — compile-verified
//
#include <hip/hip_runtime.h>
#include <hip/hip_bf16.h>

// GLIF spiking network forward pass for MI455X (gfx1250, wave32).
// Model constants fixed by the reference:
#define GLIF_D   5      // spike delay ring depth (only visible via rec_cols range)
#define GLIF_R   4      // synapse channels per neuron
#define GLIF_DT  1.0f

#define EDGE_TPB    256
#define TILE_EDGES  1024   // EDGE_TPB lanes x 4 edges (16B) per lane

// ---------------------------------------------------------------------------
// gfx1250 async-to-LDS feature detection (device pass only).
// ---------------------------------------------------------------------------
#if defined(__HIP_DEVICE_COMPILE__)
  #if __has_builtin(__builtin_amdgcn_global_load_async_to_lds_b128) && \
      __has_builtin(__builtin_amdgcn_s_wait_asynccnt)
    #define GLIF_ASYNC_LDS 1
    #pragma message("GLIF: gfx1250 async-to-LDS path ENABLED")
  #else
    #define GLIF_ASYNC_LDS 0
    #pragma message("GLIF: async-to-LDS builtins NOT found; direct-load path")
  #endif
#else
  #define GLIF_ASYNC_LDS 0
#endif

#if GLIF_ASYNC_LDS
// Exact pointer types the builtin expects (from the round-2 diagnostic):
// src = AS(1) pointer to 16B int vector, dst = AS(3) pointer.
typedef int glif_v4i __attribute__((vector_size(16)));
typedef __attribute__((address_space(1))) glif_v4i* glif_gptr;
typedef __attribute__((address_space(3))) glif_v4i* glif_lptr;
#endif

__device__ __forceinline__ void atomic_add_f32_agent(float* p, float v) {
    // Relaxed agent-scope fadd -> intended lowering: global_atomic_add_f32.
    __hip_atomic_fetch_add(p, v, __ATOMIC_RELAXED, __HIP_MEMORY_SCOPE_AGENT);
}

// ---------------------------------------------------------------------------
// Lowering probe (FIRST in file so the disasm snippet shows it; never
// launched): must show a single native global_atomic_add_f32, not a CAS loop.
// ---------------------------------------------------------------------------
__global__ void glif_atomic_probe(float* p, float v) {
    atomic_add_f32_agent(p, v);
}

// ---------------------------------------------------------------------------
// Edge pass:  i_rec[rows[e]] += w[e] * z[cols[e]]
// z is binary {0,1}; the contribution is exactly w when z==1, so the multiply
// is dropped and the atomic is guarded. A ring-occupancy counter (exact,
// maintained by the neuron pass) lets the whole kernel early-out uniformly
// when no spike is anywhere in the D-deep delay ring (always true at t=0 and
// during ramp-up).
//
// Async path: double-buffered LDS staging of the (w, rows, cols) streams via
// global_load_async_to_lds_b128. Each lane stages its own 16B per array, so
// the per-wave ASYNCcnt wait is sufficient (no cross-wave LDS sharing, no
// barriers). The async engine streams the next 1024-edge tile while the wave
// performs the random z-gather + L2 atomics for the current tile.
// ---------------------------------------------------------------------------
__global__ __launch_bounds__(EDGE_TPB) void glif_edges(
    const float* __restrict__ w, const int* __restrict__ rows,
    const int* __restrict__ cols, const float* __restrict__ z,
    float* __restrict__ i_rec, const int* __restrict__ ring_cnt, int nnz)
{
    if (ring_cnt[0] == 0) return;   // no spikes in the delay ring -> nothing to add

#if GLIF_ASYNC_LDS
    __shared__ __align__(16) float lds_w[2][TILE_EDGES];
    __shared__ __align__(16) int   lds_r[2][TILE_EDGES];
    __shared__ __align__(16) int   lds_c[2][TILE_EDGES];

    const int lt     = threadIdx.x << 2;    // this lane's element slot in a tile
    const int ntiles = nnz >> 10;           // full 1024-edge tiles

    auto stage = [&](int tl, int buf) {
        const int eb = (tl << 10) + lt;     // first edge this lane stages
        __builtin_amdgcn_global_load_async_to_lds_b128(
            (glif_gptr)(w + eb),    (glif_lptr)&lds_w[buf][lt], 0, 0);
        __builtin_amdgcn_global_load_async_to_lds_b128(
            (glif_gptr)(rows + eb), (glif_lptr)&lds_r[buf][lt], 0, 0);
        __builtin_amdgcn_global_load_async_to_lds_b128(
            (glif_gptr)(cols + eb), (glif_lptr)&lds_c[buf][lt], 0, 0);
    };

    int tile = blockIdx.x;
    if (tile < ntiles) {
        stage(tile, 0);
        int buf = 0;
        for (; tile < ntiles; tile += gridDim.x) {
            const int nxt = tile + gridDim.x;
            if (nxt < ntiles) {
                stage(nxt, buf ^ 1);
                __builtin_amdgcn_s_wait_asynccnt(3);  // current tile's 3 done
            } else {
                __builtin_amdgcn_s_wait_asynccnt(0);
            }
            float4 wv = *(const float4*)&lds_w[buf][lt];
            int4   rv = *(const int4*)&lds_r[buf][lt];
            int4   cv = *(const int4*)&lds_c[buf][lt];
            float z0 = z[cv.x], z1 = z[cv.y], z2 = z[cv.z], z3 = z[cv.w];
            if (z0 != 0.0f) atomic_add_f32_agent(&i_rec[rv.x], wv.x);
            if (z1 != 0.0f) atomic_add_f32_agent(&i_rec[rv.y], wv.y);
            if (z2 != 0.0f) atomic_add_f32_agent(&i_rec[rv.z], wv.z);
            if (z3 != 0.0f) atomic_add_f32_agent(&i_rec[rv.w], wv.w);
            buf ^= 1;
        }
    }
    // Tail (< TILE_EDGES edges), handled by block 0 with direct loads.
    if (blockIdx.x == 0) {
        for (int e = (ntiles << 10) + threadIdx.x; e < nnz; e += EDGE_TPB) {
            if (z[cols[e]] != 0.0f) atomic_add_f32_agent(&i_rec[rows[e]], w[e]);
        }
    }
#else
    // Direct-load fallback: 4 edges/thread via 16B loads + prefetch.
    int tid    = blockIdx.x * blockDim.x + threadIdx.x;
    int stride = gridDim.x * blockDim.x;
    int nvec   = nnz >> 2;

    for (int i = tid; i < nvec; i += stride) {
        int base = i << 2;
        int nb = base + (stride << 2);
        if (nb + 3 < nnz) {
            __builtin_prefetch(&w[nb], 0, 1);
            __builtin_prefetch(&rows[nb], 0, 1);
            __builtin_prefetch(&cols[nb], 0, 1);
        }
        float4 wv = *(const float4*)(w + base);
        int4   rv = *(const int4*)(rows + base);
        int4   cv = *(const int4*)(cols + base);
        if (z[cv.x] != 0.0f) atomic_add_f32_agent(&i_rec[rv.x], wv.x);
        if (z[cv.y] != 0.0f) atomic_add_f32_agent(&i_rec[rv.y], wv.y);
        if (z[cv.z] != 0.0f) atomic_add_f32_agent(&i_rec[rv.z], wv.z);
        if (z[cv.w] != 0.0f) atomic_add_f32_agent(&i_rec[rv.w], wv.w);
    }
    if (tid == 0) {
        for (int e = nvec << 2; e < nnz; ++e) {
            if (z[cols[e]] != 0.0f) atomic_add_f32_agent(&i_rec[rows[e]], w[e]);
        }
    }
#endif
}

// ---------------------------------------------------------------------------
// Neuron pass: one thread per neuron. Consumes i_rec (then zeroes it for the
// next step), updates all state, emits z/v outputs, shifts the delay ring
// into the ping-pong destination buffer, and maintains the exact ring
// spike-occupancy counter (+1 per new spike, -1 per spike aging out).
// ---------------------------------------------------------------------------
__global__ __launch_bounds__(256) void glif_neuron(
    const float* __restrict__ x_t,          // inputs + t*N*R
    float* __restrict__ i_rec,              // read + reset to 0
    const float* __restrict__ v_reset, const float* __restrict__ decay,
    const float* __restrict__ current_factor, const float* __restrict__ t_ref,
    const float* __restrict__ asc_amps,     // (N,2)
    const float* __restrict__ kparam,       // (N,2)
    const float* __restrict__ syn_decay,    // (N,R)
    const float* __restrict__ psc_initial,  // (N,R)
    const float* __restrict__ param_g, const float* __restrict__ e_l,
    const float* __restrict__ v_th, const float* __restrict__ vscale,
    const float* __restrict__ voffset,
    const float* __restrict__ z_cur, float* __restrict__ z_next,
    float* __restrict__ v, float* __restrict__ rbuf,
    float* __restrict__ asc1, float* __restrict__ asc2,
    float* __restrict__ psc_rise, float* __restrict__ psc,
    int* __restrict__ ring_cnt,
    float* __restrict__ out_z,              // d_out + t*N
    float* __restrict__ out_v,              // d_out + T*N + t*N
    int N)
{
    int n = blockIdx.x * blockDim.x + threadIdx.x;
    if (n >= N) return;

    const int b4 = n * 4;
    const float prev_z = z_cur[n];

    float4 sd = *(const float4*)(syn_decay + b4);
    float4 pi = *(const float4*)(psc_initial + b4);
    float4 pr = *(const float4*)(psc_rise + b4);
    float4 pc = *(const float4*)(psc + b4);
    float4 ir = *(const float4*)(i_rec + b4);
    float4 xt = *(const float4*)(x_t + b4);

    // input current uses OLD psc
    float input_current = pc.x + pc.y + pc.z + pc.w;

    // rec_in = i_rec + x_t ; synapse state updates (DT == 1)
    float4 npr, npc;
    npr.x = pr.x * sd.x + (ir.x + xt.x) * pi.x;
    npr.y = pr.y * sd.y + (ir.y + xt.y) * pi.y;
    npr.z = pr.z * sd.z + (ir.z + xt.z) * pi.z;
    npr.w = pr.w * sd.w + (ir.w + xt.w) * pi.w;
    npc.x = pc.x * sd.x + GLIF_DT * sd.x * pr.x;
    npc.y = pc.y * sd.y + GLIF_DT * sd.y * pr.y;
    npc.z = pc.z * sd.z + GLIF_DT * sd.z * pr.z;
    npc.w = pc.w * sd.w + GLIF_DT * sd.w * pr.w;

    // refractory
    float new_r = fmaxf(rbuf[n] + prev_z * t_ref[n] - GLIF_DT, 0.0f);

    // after-spike currents (use OLD asc values in the voltage update)
    float2 amps = *(const float2*)(asc_amps + n * 2);
    float2 kk   = *(const float2*)(kparam + n * 2);
    float a1 = asc1[n], a2 = asc2[n];
    float na1 = expf(-GLIF_DT * kk.x) * a1 + prev_z * amps.x;
    float na2 = expf(-GLIF_DT * kk.y) * a2 + prev_z * amps.y;

    // membrane voltage
    float vth = v_th[n], el = e_l[n];
    float reset_current = prev_z * (v_reset[n] - vth);
    float nv = decay[n] * v[n]
             + current_factor[n] * (input_current + a1 + a2 + param_g[n] * el)
             + reset_current;

    // spike with surrogate forward = heaviside, gated by refractory
    float v_sc = (nv - vth) / (vth - el);
    float nz = (v_sc > 0.0f) ? 1.0f : 0.0f;
    if (new_r > 0.0f) nz = 0.0f;

    // state writeback
    *(float4*)(psc_rise + b4) = npr;
    *(float4*)(psc + b4)      = npc;
    *(float4*)(i_rec + b4)    = make_float4(0.0f, 0.0f, 0.0f, 0.0f); // for next step
    v[n]    = nv;
    rbuf[n] = new_r;
    asc1[n] = na1;
    asc2[n] = na2;

    // delay-ring shift: new row 0 = nz, rows 1..D-1 = old rows 0..D-2.
    // Old row D-1 falls out of the ring.
    z_next[n] = nz;
#pragma unroll
    for (int d = 1; d < GLIF_D; ++d)
        z_next[d * N + n] = z_cur[(d - 1) * N + n];

    // exact ring occupancy update
    int delta = 0;
    if (nz != 0.0f) delta += 1;
    if (z_cur[(GLIF_D - 1) * N + n] != 0.0f) delta -= 1;
    if (delta != 0) atomicAdd(ring_cnt, delta);

    // outputs
    out_z[n] = nz;
    out_v[n] = nv * vscale[n] + voffset[n];
}

// ---------------------------------------------------------------------------
// Init: zero spike ring (current buffer), i_rec, psc state; v = v_reset.
// ---------------------------------------------------------------------------
__global__ __launch_bounds__(256) void glif_init(
    float* __restrict__ z_cur, float* __restrict__ i_rec,
    float* __restrict__ psc_rise, float* __restrict__ psc,
    float* __restrict__ v, float* __restrict__ r,
    float* __restrict__ asc1, float* __restrict__ asc2,
    int* __restrict__ ring_cnt,
    const float* __restrict__ v_reset, int N)
{
    int i = blockIdx.x * blockDim.x + threadIdx.x;
    int DN = GLIF_D * N;
    int NR = N * GLIF_R;
    if (i < DN) z_cur[i] = 0.0f;
    if (i < NR) { i_rec[i] = 0.0f; psc_rise[i] = 0.0f; psc[i] = 0.0f; }
    if (i < N)  { v[i] = v_reset[i]; r[i] = 0.0f; asc1[i] = 0.0f; asc2[i] = 0.0f; }
    if (i == 0) ring_cnt[0] = 0;
}

extern "C" void kernel_launch(void* const* d_in, const int* in_sizes, int n_in,
                              void* d_out, int out_size, void* d_ws, size_t ws_size,
                              hipStream_t stream) {
    const float* inputs    = (const float*)d_in[0];
    const float* w         = (const float*)d_in[1];
    const float* v_reset   = (const float*)d_in[2];
    const float* decay     = (const float*)d_in[3];
    const float* cur_fac   = (const float*)d_in[4];
    const float* t_ref     = (const float*)d_in[5];
    const float* asc_amps  = (const float*)d_in[6];
    const float* kparam    = (const float*)d_in[7];
    const float* syn_decay = (const float*)d_in[8];
    const float* psc_init  = (const float*)d_in[9];
    const float* param_g   = (const float*)d_in[10];
    const float* e_l       = (const float*)d_in[11];
    const float* v_th      = (const float*)d_in[12];
    const float* vscale    = (const float*)d_in[13];
    const float* voffset   = (const float*)d_in[14];
    const int*   rows      = (const int*)d_in[15];
    const int*   cols      = (const int*)d_in[16];

    const int N   = in_sizes[2];                 // 100000
    const int NNZ = in_sizes[1];                 // 5,000,000
    const int R   = in_sizes[8] / N;             // 4 (kernels hardcode 4)
    const int T   = in_sizes[0] / (N * R);       // 25 (B == 1)
    const int D   = GLIF_D;                      // 5

    float* out_z_base = (float*)d_out;           // (T, N)
    float* out_v_base = (float*)d_out + (size_t)T * N;

    // Workspace layout (floats; every sub-buffer stays 16B aligned)
    float* ws  = (float*)d_ws;
    size_t DN  = (size_t)D * N;
    size_t NR  = (size_t)N * R;
    float* z_a      = ws;            ws += DN;
    float* z_b      = ws;            ws += DN;
    float* i_rec    = ws;            ws += NR;
    float* v        = ws;            ws += N;
    float* r        = ws;            ws += N;
    float* asc1     = ws;            ws += N;
    float* asc2     = ws;            ws += N;
    float* psc_rise = ws;            ws += NR;
    float* psc      = ws;            ws += NR;
    int*   ring_cnt = (int*)ws;      ws += 4;

    const int TPB = 256;
    const int init_blocks   = (int)((DN + TPB - 1) / TPB);
    const int neuron_blocks = (N + TPB - 1) / TPB;
    const int edge_blocks   = 1024;  // grid-stride over 1024-edge tiles

    glif_init<<<init_blocks, TPB, 0, stream>>>(
        z_a, i_rec, psc_rise, psc, v, r, asc1, asc2, ring_cnt, v_reset, N);

    float* z_cur  = z_a;
    float* z_next = z_b;
    for (int t = 0; t < T; ++t) {
        glif_edges<<<edge_blocks, EDGE_TPB, 0, stream>>>(
            w, rows, cols, z_cur, i_rec, ring_cnt, NNZ);
        glif_neuron<<<neuron_blocks, TPB, 0, stream>>>(
            inputs + (size_t)t * N * R, i_rec,
            v_reset, decay, cur_fac, t_ref, asc_amps, kparam,
            syn_decay, psc_init, param_g, e_l, v_th, vscale, voffset,
            z_cur, z_next, v, r, asc1, asc2, psc_rise, psc, ring_cnt,
            out_z_base + (size_t)t * N, out_v_base + (size_t)t * N, N);
        float* tmp = z_cur; z_cur = z_next; z_next = tmp;
    }
}